// MOERouter_80951543595521
// MI455X (gfx1250) — compile-verified
//
#include <hip/hip_runtime.h>

typedef __attribute__((ext_vector_type(2))) float v2f;
typedef __attribute__((ext_vector_type(8))) float v8f;

#define DIM 4096
#define NEXP 64
#define TOK_PER_BLK 64
#define ROWSTRIDE 68   // 64 + 4 pad: LDS bank-conflict avoidance

// ---------------------------------------------------------------------------
// ws accumulator init (graph-replay safe: reset every launch)
// ---------------------------------------------------------------------------
__global__ void moe_init(float* ws) {
    int i = threadIdx.x;
    if (i < 2 * NEXP) ws[i] = 0.0f;
}

// ---------------------------------------------------------------------------
// Fused: gate GEMM (WMMA f32 16x16x4) -> softmax -> top-2 -> dense scatter
//        + per-block expert score-sums / top1 counts
// ---------------------------------------------------------------------------
__global__ __launch_bounds__(128) void moe_router(
        const float* __restrict__ x,      // [nTokens, 4096]
        const float* __restrict__ Wg,     // [4096, 64]
        float* __restrict__ out,          // importance | indicator | l_aux
        float* __restrict__ gsum,         // [64] global score sums
        float* __restrict__ gcnt,         // [64] global top1 counts
        int nTokens) {
    __shared__ float sc[TOK_PER_BLK * ROWSTRIDE];  // logits -> scores
    __shared__ float cntS[NEXP];

    const int tid  = threadIdx.x;
    const int wave = tid >> 5;          // 4 waves / block (wave32)
    const int lane = tid & 31;
    const int lo   = lane & 15;
    const int hi   = lane >> 4;

    const int tileBase = blockIdx.x * TOK_PER_BLK + wave * 16;

    // A layout (f32 16x4, MxK): lanes 0-15 hold row M=lane, K={k,k+1};
    //                           lanes 16-31 hold row M=lane-16, K={k+2,k+3}
    const float* Aptr = x + (size_t)(tileBase + lo) * DIM + 2 * hi;
    // B layout (f32 4x16, KxN): lanes 0-15 rows K={k,k+1}; lanes 16-31 K={k+2,k+3}
    const float* Bptr = Wg + (2 * hi) * NEXP + lo;

    v8f acc0 = {}, acc1 = {}, acc2 = {}, acc3 = {};

    #pragma unroll 4
    for (int k = 0; k < DIM; k += 4) {
        v2f a = *(const v2f*)(Aptr + k);          // 8B contiguous per lane
        const float* bp = Bptr + k * NEXP;
        v2f b0 = { bp[0],  bp[NEXP + 0]  };
        v2f b1 = { bp[16], bp[NEXP + 16] };
        v2f b2 = { bp[32], bp[NEXP + 32] };
        v2f b3 = { bp[48], bp[NEXP + 48] };
        acc0 = __builtin_amdgcn_wmma_f32_16x16x4_f32(false, a, false, b0, (short)0, acc0, false, false);
        acc1 = __builtin_amdgcn_wmma_f32_16x16x4_f32(false, a, false, b1, (short)0, acc1, false, false);
        acc2 = __builtin_amdgcn_wmma_f32_16x16x4_f32(false, a, false, b2, (short)0, acc2, false, false);
        acc3 = __builtin_amdgcn_wmma_f32_16x16x4_f32(false, a, false, b3, (short)0, acc3, false, false);
    }

    // C layout (f32 16x16): VGPR r -> M=r (lanes 0-15), M=r+8 (lanes 16-31), N=lane&15
    {
        const int rbase = wave * 16 + 8 * hi;
        #pragma unroll
        for (int r = 0; r < 8; ++r) {
            float* dst = &sc[(rbase + r) * ROWSTRIDE + lo];
            dst[0]  = acc0[r];
            dst[16] = acc1[r];
            dst[32] = acc2[r];
            dst[48] = acc3[r];
        }
    }
    if (tid < NEXP) cntS[tid] = 0.0f;
    __syncthreads();

    // One thread per token: softmax over 64, top-2, coalesced dense scatter
    if (tid < TOK_PER_BLK) {
        float* row = &sc[tid * ROWSTRIDE];
        float m = row[0];
        #pragma unroll
        for (int e = 1; e < NEXP; ++e) m = fmaxf(m, row[e]);
        float s = 0.0f;
        #pragma unroll
        for (int e = 0; e < NEXP; ++e) { float p = __expf(row[e] - m); row[e] = p; s += p; }
        const float inv = 1.0f / s;

        float v1 = -1.0f, v2 = -1.0f;
        int   i1 = 0,     i2 = 0;
        #pragma unroll
        for (int e = 0; e < NEXP; ++e) {
            float p = row[e] * inv;
            row[e] = p;                       // keep normalized scores for column sums
            if (p > v1)      { v2 = v1; i2 = i1; v1 = p; i1 = e; }
            else if (p > v2) { v2 = p;  i2 = e; }
        }

        const size_t tg  = (size_t)blockIdx.x * TOK_PER_BLK + tid;
        float* imp = out;
        float* ind = out + (size_t)NEXP * nTokens;
        #pragma unroll
        for (int e = 0; e < NEXP; ++e) {
            float v = 0.0f, w = 0.0f;
            if (e == i1)      { v = v1; w = 1.0f; }
            else if (e == i2) { v = v2; w = 1.0f; }
            imp[(size_t)e * nTokens + tg] = v;   // consecutive tids -> coalesced
            ind[(size_t)e * nTokens + tg] = w;
        }
        atomicAdd(&cntS[i1], 1.0f);
    }
    __syncthreads();

    // Per-expert column sums over the block's 64 tokens; one atomic per expert
    if (tid < NEXP) {
        float s = 0.0f;
        #pragma unroll 8
        for (int r = 0; r < TOK_PER_BLK; ++r) s += sc[r * ROWSTRIDE + tid];
        atomicAdd(&gsum[tid], s);
        atomicAdd(&gcnt[tid], cntS[tid]);
    }
}

// ---------------------------------------------------------------------------
// l_aux = E^2 * sum_e( mean_scores[e] * frac_top1[e] )
// ---------------------------------------------------------------------------
__global__ void moe_finalize(const float* __restrict__ gsum,
                             const float* __restrict__ gcnt,
                             float* __restrict__ out, int nTokens) {
    __shared__ float red[NEXP];
    int e = threadIdx.x;
    red[e] = gsum[e] * gcnt[e];
    __syncthreads();
    if (e == 0) {
        float s = 0.0f;
        for (int i = 0; i < NEXP; ++i) s += red[i];
        double scale = (double)NEXP * (double)NEXP /
                       ((double)nTokens * (double)nTokens);
        out[(size_t)2 * NEXP * nTokens] = (float)(s * scale);
    }
}

extern "C" void kernel_launch(void* const* d_in, const int* in_sizes, int n_in,
                              void* d_out, int out_size, void* d_ws, size_t ws_size,
                              hipStream_t stream) {
    const float* x  = (const float*)d_in[0];   // [4, 8192, 4096] f32
    const float* Wg = (const float*)d_in[1];   // [4096, 64] f32
    float* out = (float*)d_out;

    const int nTokens = in_sizes[0] / DIM;     // 32768
    float* gsum = (float*)d_ws;
    float* gcnt = gsum + NEXP;

    moe_init<<<1, 128, 0, stream>>>(gsum);
    const int grid = nTokens / TOK_PER_BLK;    // 512
    moe_router<<<grid, 128, 0, stream>>>(x, Wg, out, gsum, gcnt, nTokens);
    moe_finalize<<<1, NEXP, 0, stream>>>(gsum, gcnt, out, nTokens);
}